// twins_svt_encode_74062416053195
// MI455X (gfx1250) — compile-verified
//
#include <hip/hip_runtime.h>
#include <hip/hip_bf16.h>
#include <math.h>

typedef __attribute__((ext_vector_type(2))) float v2f;
typedef __attribute__((ext_vector_type(8))) float v8f;

// D = A(16x4) * B(4x16) + C, fp32 WMMA (CDNA5 V_WMMA_F32_16X16X4_F32)
__device__ __forceinline__ v8f wmma4(v2f a, v2f b, v8f c) {
  return __builtin_amdgcn_wmma_f32_16x16x4_f32(false, a, false, b, (short)0, c, false, false);
}

// ---------- sizes ----------
#define NB   64          // 4 batch * 4 heads * 4 windows
#define NN   4096        // 64*64 spatial per window
#define CCH  32          // head dim
#define M100 100
#define MPAD 112
#define SIMS ((size_t)MPAD * NN)   // 458752 per b

// workspace offsets (in floats)
#define OFF_XQ   ((size_t)0)
#define OFF_VAL  ((size_t)8388608)          // also v_n (in place), later xtn (alias)
#define OFF_FEAT ((size_t)16777216)
#define OFF_CEN  ((size_t)25165824)         // 64*3200 centers [c][pq] flat
#define OFF_CC   (OFF_CEN + 204800)         // conv_c output [o][pq] flat
#define OFF_CN   (OFF_CC + 204800)          // 64*3584 normalized centers rows [m][c]
#define OFF_CF   (OFF_CN + 229376)          // centers_feature [c][pq] flat
#define OFF_CTN  (OFF_CF + 204800)          // 64*3584
#define OFF_SIM  (OFF_CTN + 229376)         // 64*112*4096 ; out2 aliases first 8388608 floats
#define OFF_AM   (OFF_SIM + 29360128)       // int 262144
#define OFF_SV   (OFF_AM + 262144)
#define OFF_CNT  (OFF_SV + 262144)          // 64*100
#define OFF_OUTC (OFF_CNT + 6400)           // 64*3584

// ---------------- Kernel 1: fused conv1/conv_v/conv_f (1x1 convs as GEMM) ----------------
// out[b][o][n] = sum_c W[o][c]*x[b][n][c] + bias[o] ; x: [4,16384,128]
__global__ void k_proj(const float* __restrict__ x,
                       const float* __restrict__ w1, const float* __restrict__ b1,
                       const float* __restrict__ wv, const float* __restrict__ bv,
                       const float* __restrict__ wf, const float* __restrict__ bf,
                       float* __restrict__ xq, float* __restrict__ val, float* __restrict__ feat) {
  int lane = threadIdx.x & 31, wave = threadIdx.x >> 5;
  int nt = blockIdx.x * 4 + wave;      // 0..1023
  int o0 = blockIdx.y * 16;            // 0..7 tiles
  int b  = blockIdx.z;                 // 0..3
  int half = lane >> 4, lm = lane & 15;
  int n0 = nt * 16;
  const float* xb = x + (size_t)b * 16384 * 128;
  v8f a0 = {}, a1 = {}, a2 = {};
#pragma unroll 4
  for (int k = 0; k < 128; k += 4) {
    int kk = k + 2 * half;
    v2f wa = *reinterpret_cast<const v2f*>(w1 + (o0 + lm) * 128 + kk);
    v2f wb = *reinterpret_cast<const v2f*>(wv + (o0 + lm) * 128 + kk);
    v2f wc = *reinterpret_cast<const v2f*>(wf + (o0 + lm) * 128 + kk);
    v2f bbv = *reinterpret_cast<const v2f*>(xb + (size_t)(n0 + lm) * 128 + kk);
    a0 = wmma4(wa, bbv, a0);
    a1 = wmma4(wb, bbv, a1);
    a2 = wmma4(wc, bbv, a2);
  }
#pragma unroll
  for (int r = 0; r < 8; ++r) {
    int o = o0 + r + half * 8;
    size_t idx = ((size_t)(b * 128 + o)) * 16384 + n0 + lm;
    xq[idx]   = a0[r] + b1[o];
    val[idx]  = a1[r] + bv[o];
    feat[idx] = a2[r] + bf[o];
  }
}

// ---------------- Kernel 2: adaptive avg pool 64x64 -> 10x10 for xq and feature ----------------
__global__ void k_pool(const float* __restrict__ xq, const float* __restrict__ feat,
                       float* __restrict__ cen, float* __restrict__ cf) {
  int t = blockIdx.x * blockDim.x + threadIdx.x;
  if (t >= NB * CCH * 100) return;
  int pq = t % 100, c = (t / 100) % CCH, b = t / (CCH * 100);
  int p = pq / 10, q = pq % 10;
  int sw = (p * 64) / 10, ew = ((p + 1) * 64 + 9) / 10;
  int sh = (q * 64) / 10, eh = ((q + 1) * 64 + 9) / 10;
  const float* s1 = xq + (size_t)b * 131072 + c * 4096;
  const float* s2 = feat + (size_t)b * 131072 + c * 4096;
  float acc1 = 0.f, acc2 = 0.f;
  for (int w = sw; w < ew; ++w)
    for (int h = sh; h < eh; ++h) { acc1 += s1[w * 64 + h]; acc2 += s2[w * 64 + h]; }
  float inv = 1.0f / (float)((ew - sw) * (eh - sh));
  cen[(size_t)b * 3200 + c * 100 + pq] = acc1 * inv;
  cf [(size_t)b * 3200 + c * 100 + pq] = acc2 * inv;
}

// ---------------- Kernel 3: l2-normalize contiguous rows of 32 (v_n, in place) ----------------
__global__ void k_norm_rows(float* __restrict__ buf, int nrows) {
  int lane = threadIdx.x & 31;
  int row = (int)((blockIdx.x * blockDim.x + threadIdx.x) >> 5);
  if (row >= nrows) return;
  float v = buf[(size_t)row * 32 + lane];
  float s = v * v;
#pragma unroll
  for (int d = 16; d >= 1; d >>= 1) s += __shfl_xor(s, d, 32);
  buf[(size_t)row * 32 + lane] = v / fmaxf(sqrtf(s), 1e-12f);
}

// ---------------- Kernel 4: conv_c 32->32 on centers [c][pq] ----------------
__global__ void k_convc(const float* __restrict__ cen, const float* __restrict__ wc,
                        const float* __restrict__ bc, float* __restrict__ cc) {
  int lane = threadIdx.x & 31, wave = threadIdx.x >> 5;
  int tile = blockIdx.x * 4 + wave;           // 7*2*64 = 896
  if (tile >= 896) return;
  int pt = tile % 7, ot = (tile / 7) % 2, b = tile / 14;
  int half = lane >> 4, lm = lane & 15;
  int pq0 = pt * 16, o0 = ot * 16;
  int pqv = pq0 + lm;
  int pqc = pqv < 100 ? pqv : 99;
  const float* cb = cen + (size_t)b * 3200;
  v8f acc = {};
#pragma unroll
  for (int k = 0; k < 32; k += 4) {
    int kk = k + 2 * half;
    v2f a = *reinterpret_cast<const v2f*>(wc + (o0 + lm) * 32 + kk);
    float bx = cb[kk * 100 + pqc];
    float by = cb[(kk + 1) * 100 + pqc];
    v2f bb; bb.x = (pqv < 100) ? bx : 0.f; bb.y = (pqv < 100) ? by : 0.f;
    acc = wmma4(a, bb, acc);
  }
#pragma unroll
  for (int r = 0; r < 8; ++r) {
    int o = o0 + r + half * 8;
    if (pqv < 100) cc[(size_t)b * 3200 + o * 100 + pqv] = acc[r] + bc[o];
  }
}

// ---------------- Kernel 5: normalize conv_c result as rows [m][c] -> cn (pad to 112) ----------
__global__ void k_norm_cn(const float* __restrict__ cc, float* __restrict__ cn) {
  int lane = threadIdx.x & 31;
  int row = (int)((blockIdx.x * blockDim.x + threadIdx.x) >> 5);
  if (row >= NB * MPAD) return;
  int m = row % MPAD, b = row / MPAD;
  float out = 0.f;
  if (m < M100) {
    float v = cc[(size_t)b * 3200 + m * 32 + lane];
    float s = v * v;
#pragma unroll
    for (int d = 16; d >= 1; d >>= 1) s += __shfl_xor(s, d, 32);
    out = v / fmaxf(sqrtf(s), 1e-12f);
  }
  cn[(size_t)b * 3584 + m * 32 + lane] = out;
}

// ---------------- Kernel 6: sim = cn (rows[m][32]) x vn^T (rows[n][32]) -> [112][4096] --------
__global__ void k_simgemm(const float* __restrict__ cn, const float* __restrict__ vn,
                          float* __restrict__ sim) {
  int lane = threadIdx.x & 31, wave = threadIdx.x >> 5;
  int nt = blockIdx.x * 4 + wave;   // 0..255
  int m0 = blockIdx.y * 16;         // 0..6 tiles
  int b  = blockIdx.z;
  int half = lane >> 4, lm = lane & 15;
  int n0 = nt * 16;
  const float* A = cn + (size_t)b * 3584;
  const float* B = vn + (size_t)b * 131072;
  v8f acc = {};
#pragma unroll
  for (int k = 0; k < 32; k += 4) {
    int kk = k + 2 * half;
    v2f a = *reinterpret_cast<const v2f*>(A + (m0 + lm) * 32 + kk);
    v2f bb = *reinterpret_cast<const v2f*>(B + (size_t)(n0 + lm) * 32 + kk);
    acc = wmma4(a, bb, acc);
  }
  float* S = sim + (size_t)b * SIMS;
#pragma unroll
  for (int r = 0; r < 8; ++r) {
    int m = m0 + r + half * 8;
    S[(size_t)m * 4096 + n0 + lm] = acc[r];
  }
}

// ---------------- Kernel 7: softmax over the m axis (100 rows) per column n -------------------
__global__ void k_softmax(float* __restrict__ sim) {
  int t = blockIdx.x * blockDim.x + threadIdx.x;
  if (t >= NB * NN) return;
  int n = t & 4095, b = t >> 12;
  float* S = sim + (size_t)b * SIMS + n;
  float v[M100];
  float mx = -3.4e38f;
#pragma unroll
  for (int m = 0; m < M100; ++m) { v[m] = S[(size_t)m * 4096]; mx = fmaxf(mx, v[m]); }
  float s = 0.f;
#pragma unroll
  for (int m = 0; m < M100; ++m) { v[m] = expf(v[m] - mx); s += v[m]; }
  float inv = 1.0f / s;
#pragma unroll
  for (int m = 0; m < M100; ++m) S[(size_t)m * 4096] = v[m] * inv;
}

// ---------------- Kernel 8: centers = sim x feature  ([112]x[4096]) . ([4096]x[32]) -----------
// k_outgemm below repeats this GEMM loop with its own epilogue
__global__ void k_cgemm(const float* __restrict__ sim, const float* __restrict__ feat,
                        float* __restrict__ cen) {
  int lane = threadIdx.x & 31, wave = threadIdx.x >> 5;
  int tile = blockIdx.x * 4 + wave;   // 7*2*64
  if (tile >= 896) return;
  int ct = tile % 2, mt = (tile / 2) % 7, b = tile / 14;
  int half = lane >> 4, lm = lane & 15;
  int m0 = mt * 16, c0 = ct * 16;
  const float* A = sim + (size_t)b * SIMS;
  const float* F = feat + (size_t)b * 131072;
  v8f accs[4] = {v8f{}, v8f{}, v8f{}, v8f{}};
  for (int k = 0; k < 4096; k += 16) {
#pragma unroll
    for (int u = 0; u < 4; ++u) {
      int kk = k + u * 4 + 2 * half;
      v2f a = *reinterpret_cast<const v2f*>(A + (size_t)(m0 + lm) * 4096 + kk);
      v2f bb; bb.x = F[(size_t)kk * 32 + c0 + lm]; bb.y = F[(size_t)(kk + 1) * 32 + c0 + lm];
      accs[u] = wmma4(a, bb, accs[u]);
    }
  }
#pragma unroll
  for (int r = 0; r < 8; ++r) {
    int m = m0 + r + half * 8;
    if (m < M100) {
      float v = (accs[0][r] + accs[1][r]) + (accs[2][r] + accs[3][r]);
      cen[(size_t)b * 3200 + m * 32 + c0 + lm] = v;
    }
  }
}

// ---------------- Kernel 9: ctn = l2n of centers-transpose rows; pad to 112 -------------------
__global__ void k_norm_ct(const float* __restrict__ cen, float* __restrict__ ctn) {
  int lane = threadIdx.x & 31;
  int row = (int)((blockIdx.x * blockDim.x + threadIdx.x) >> 5);
  if (row >= NB * MPAD) return;
  int m = row % MPAD, b = row / MPAD;
  float out = 0.f;
  if (m < M100) {
    float v = cen[(size_t)b * 3200 + lane * 100 + m];
    float s = v * v;
#pragma unroll
    for (int d = 16; d >= 1; d >>= 1) s += __shfl_xor(s, d, 32);
    out = v / fmaxf(sqrtf(s), 1e-12f);
  }
  ctn[(size_t)b * 3584 + m * 32 + lane] = out;
}

// ---------------- Kernel 10: xtn[n][c] = l2n over c of xq[c][n] ------------------------------
__global__ void k_norm_xt(const float* __restrict__ xq, float* __restrict__ xtn) {
  int lane = threadIdx.x & 31;
  int row = (int)((blockIdx.x * blockDim.x + threadIdx.x) >> 5);
  if (row >= NB * NN) return;
  int n = row & 4095, b = row >> 12;
  float v = xq[(size_t)b * 131072 + (size_t)lane * 4096 + n];
  float s = v * v;
#pragma unroll
  for (int d = 16; d >= 1; d >>= 1) s += __shfl_xor(s, d, 32);
  xtn[(size_t)b * 131072 + (size_t)n * 32 + lane] = v / fmaxf(sqrtf(s), 1e-12f);
}

// ---------------- Kernel 11: final sim = sigmoid(beta + alpha * ctn x xtn^T) ------------------
__global__ void k_fsim(const float* __restrict__ ctn, const float* __restrict__ xtn,
                       const float* __restrict__ alpha, const float* __restrict__ beta,
                       float* __restrict__ sim) {
  int lane = threadIdx.x & 31, wave = threadIdx.x >> 5;
  int nt = blockIdx.x * 4 + wave;
  int m0 = blockIdx.y * 16;
  int b  = blockIdx.z;
  int half = lane >> 4, lm = lane & 15;
  int n0 = nt * 16;
  const float* A = ctn + (size_t)b * 3584;
  const float* B = xtn + (size_t)b * 131072;
  v8f acc = {};
#pragma unroll
  for (int k = 0; k < 32; k += 4) {
    int kk = k + 2 * half;
    v2f a = *reinterpret_cast<const v2f*>(A + (m0 + lm) * 32 + kk);
    v2f bb = *reinterpret_cast<const v2f*>(B + (size_t)(n0 + lm) * 32 + kk);
    acc = wmma4(a, bb, acc);
  }
  float al = alpha[0], be = beta[0];
  float* S = sim + (size_t)b * SIMS;
#pragma unroll
  for (int r = 0; r < 8; ++r) {
    int m = m0 + r + half * 8;
    float d = be + al * acc[r];
    S[(size_t)m * 4096 + n0 + lm] = 1.0f / (1.0f + expf(-d));
  }
}

// ---------------- Kernel 12: per-column argmax over m, mask sim in place ----------------------
__global__ void k_argmax(float* __restrict__ sim, int* __restrict__ am, float* __restrict__ sv) {
  int t = blockIdx.x * blockDim.x + threadIdx.x;
  if (t >= NB * NN) return;
  int n = t & 4095, b = t >> 12;
  float* S = sim + (size_t)b * SIMS + n;
  float v[M100];
  float best = -3.4e38f; int bi = 0;
#pragma unroll
  for (int m = 0; m < M100; ++m) {
    v[m] = S[(size_t)m * 4096];
    if (v[m] > best) { best = v[m]; bi = m; }
  }
  am[t] = bi; sv[t] = best;
#pragma unroll
  for (int m = 0; m < M100; ++m) S[(size_t)m * 4096] = (m == bi) ? best : 0.f;
}

// ---------------- Kernel 13: counts[m] = #n assigned to m -------------------------------------
__global__ void k_counts(const int* __restrict__ am, float* __restrict__ cnt) {
  int lane = threadIdx.x & 31;
  int row = (int)((blockIdx.x * blockDim.x + threadIdx.x) >> 5);
  if (row >= NB * M100) return;
  int m = row % M100, b = row / M100;
  const int* A = am + b * 4096;
  int c = 0;
  for (int n = lane; n < 4096; n += 32) c += (A[n] == m) ? 1 : 0;
#pragma unroll
  for (int d = 16; d >= 1; d >>= 1) c += __shfl_xor(c, d, 32);
  if (lane == 0) cnt[row] = (float)c;
}

// ---------------- Kernel 14: out = (sim_masked x feature + cf) / (counts+1) -------------------
__global__ void k_outgemm(const float* __restrict__ sim, const float* __restrict__ feat,
                          const float* __restrict__ cf, const float* __restrict__ cnt,
                          float* __restrict__ outc) {
  int lane = threadIdx.x & 31, wave = threadIdx.x >> 5;
  int tile = blockIdx.x * 4 + wave;
  if (tile >= 896) return;
  int ct = tile % 2, mt = (tile / 2) % 7, b = tile / 14;
  int half = lane >> 4, lm = lane & 15;
  int m0 = mt * 16, c0 = ct * 16;
  const float* A = sim + (size_t)b * SIMS;
  const float* F = feat + (size_t)b * 131072;
  v8f accs[4] = {v8f{}, v8f{}, v8f{}, v8f{}};
  for (int k = 0; k < 4096; k += 16) {
#pragma unroll
    for (int u = 0; u < 4; ++u) {
      int kk = k + u * 4 + 2 * half;
      v2f a = *reinterpret_cast<const v2f*>(A + (size_t)(m0 + lm) * 4096 + kk);
      v2f bb; bb.x = F[(size_t)kk * 32 + c0 + lm]; bb.y = F[(size_t)(kk + 1) * 32 + c0 + lm];
      accs[u] = wmma4(a, bb, accs[u]);
    }
  }
#pragma unroll
  for (int r = 0; r < 8; ++r) {
    int m = m0 + r + half * 8;
    if (m < M100) {
      float v = (accs[0][r] + accs[1][r]) + (accs[2][r] + accs[3][r]);
      float d = cnt[(size_t)b * M100 + m] + 1.0f;
      outc[(size_t)b * 3584 + m * 32 + c0 + lm] =
          (v + cf[(size_t)b * 3200 + m * 32 + c0 + lm]) / d;
    }
  }
}

// ---------------- Kernel 15: out2[n][c] = outc[argmax(n)][c] * simval[n] ----------------------
__global__ void k_gather(const float* __restrict__ outc, const int* __restrict__ am,
                         const float* __restrict__ sv, float* __restrict__ out2) {
  size_t t = (size_t)blockIdx.x * blockDim.x + threadIdx.x;
  if (t >= (size_t)NB * NN * 32) return;
  int c = (int)(t & 31);
  int n = (int)((t >> 5) & 4095);
  int b = (int)(t >> 17);
  int a = am[b * 4096 + n];
  out2[t] = outc[(size_t)b * 3584 + a * 32 + c] * sv[b * 4096 + n];
}

// ---------------- Kernel 16: conv2 (1x1, 128->128) with transposed store ----------------------
// X2: [4][128][16384]; dout[b][s][o] = sum_C W2[o][C]*X2[b][C][s] + b2[o]
__global__ void k_conv2(const float* __restrict__ X2, const float* __restrict__ w2,
                        const float* __restrict__ b2, float* __restrict__ dout) {
  int lane = threadIdx.x & 31, wave = threadIdx.x >> 5;
  int st = blockIdx.x * 4 + wave;   // 0..1023
  int o0 = blockIdx.y * 16;         // 0..7 tiles
  int b  = blockIdx.z;              // 0..3
  int half = lane >> 4, lm = lane & 15;
  int s0 = st * 16;
  const float* X = X2 + (size_t)b * 2097152;
  v8f acc = {};
#pragma unroll 4
  for (int k = 0; k < 128; k += 4) {
    int kk = k + 2 * half;
    v2f a; a.x = X[(size_t)kk * 16384 + s0 + lm]; a.y = X[(size_t)(kk + 1) * 16384 + s0 + lm];
    v2f bb = *reinterpret_cast<const v2f*>(w2 + (o0 + lm) * 128 + kk);
    acc = wmma4(a, bb, acc);
  }
#pragma unroll
  for (int r = 0; r < 8; ++r) {
    int s = s0 + r + half * 8;
    dout[((size_t)b * 16384 + s) * 128 + o0 + lm] = acc[r] + b2[o0 + lm];
  }
}

extern "C" void kernel_launch(void* const* d_in, const int* in_sizes, int n_in,
                              void* d_out, int out_size, void* d_ws, size_t ws_size,
                              hipStream_t stream) {
  (void)in_sizes; (void)n_in; (void)out_size; (void)ws_size;
  const float* x   = (const float*)d_in[0];
  const float* w1  = (const float*)d_in[1];
  const float* b1  = (const float*)d_in[2];
  const float* wv  = (const float*)d_in[3];
  const float* bv  = (const float*)d_in[4];
  const float* wf  = (const float*)d_in[5];
  const float* bf  = (const float*)d_in[6];
  const float* wc  = (const float*)d_in[7];
  const float* bc  = (const float*)d_in[8];
  const float* w2  = (const float*)d_in[9];
  const float* b2  = (const float*)d_in[10];
  const float* alp = (const float*)d_in[11];
  const float* bet = (const float*)d_in[12];

  float* ws   = (float*)d_ws;
  float* xq   = ws + OFF_XQ;
  float* val  = ws + OFF_VAL;    // value -> v_n (in place); later reused as xtn
  float* feat = ws + OFF_FEAT;
  float* cen  = ws + OFF_CEN;
  float* cc   = ws + OFF_CC;
  float* cn   = ws + OFF_CN;
  float* cf   = ws + OFF_CF;
  float* ctn  = ws + OFF_CTN;
  float* sim  = ws + OFF_SIM;
  int*   am   = (int*)(ws + OFF_AM);
  float* sv   = ws + OFF_SV;
  float* cnt  = ws + OFF_CNT;
  float* outc = ws + OFF_OUTC;
  float* xtn  = val;             // alias (v_n dead after clustering loop)
  float* out2 = sim;             // alias (sim dead after out-GEMM)
  float* dout = (float*)d_out;

  // 1) projections
  k_proj<<<dim3(256, 8, 4), 128, 0, stream>>>(x, w1, b1, wv, bv, wf, bf, xq, val, feat);
  // 2) pooling -> centers, centers_feature
  k_pool<<<dim3(800), 256, 0, stream>>>(xq, feat, cen, cf);
  // 3) v_n = l2n(value rows)
  k_norm_rows<<<dim3(32768), 256, 0, stream>>>(val, NB * NN);
  // 4) clustering iterations
  for (int it = 0; it < 3; ++it) {
    k_convc<<<dim3(224), 128, 0, stream>>>(cen, wc, bc, cc);
    k_norm_cn<<<dim3(896), 256, 0, stream>>>(cc, cn);
    k_simgemm<<<dim3(64, 7, 64), 128, 0, stream>>>(cn, val, sim);
    k_softmax<<<dim3(1024), 256, 0, stream>>>(sim);
    k_cgemm<<<dim3(224), 128, 0, stream>>>(sim, feat, cen);
  }
  // 5) normalized ct / xt
  k_norm_ct<<<dim3(896), 256, 0, stream>>>(cen, ctn);
  k_norm_xt<<<dim3(32768), 256, 0, stream>>>(xq, xtn);
  // 6) final sim with sigmoid
  k_fsim<<<dim3(64, 7, 64), 128, 0, stream>>>(ctn, xtn, alp, bet, sim);
  // 7) argmax + mask, counts
  k_argmax<<<dim3(1024), 256, 0, stream>>>(sim, am, sv);
  k_counts<<<dim3(800), 256, 0, stream>>>(am, cnt);
  // 8) out = (sim_masked @ feature + cf)/(counts+1)
  k_outgemm<<<dim3(224), 128, 0, stream>>>(sim, feat, cf, cnt, outc);
  // 9) gather out2 (aliases sim region; sim no longer needed)
  k_gather<<<dim3(32768), 256, 0, stream>>>(outc, am, sv, out2);
  // 10) conv2 + transposed store to d_out
  k_conv2<<<dim3(256, 8, 4), 128, 0, stream>>>(out2, w2, b2, dout);
}